// MultiHeadedAttention_spatial_29807073034257
// MI455X (gfx1250) — compile-verified
//
#include <hip/hip_runtime.h>

// ---------------------------------------------------------------------------
// MultiHeadedAttention_spatial on MI455X (gfx1250, wave32).
//
// The reference's softmax is over a singleton axis => p_attn == 1 identically,
// so mask / Wk / bk / scores are numerically dead. Effective computation:
//   q = query @ Wq^T + bq
//   v = q     @ Wv^T + bv
//   w = v     @ Wo^T + bo          (per-node, computed ONCE, not per neighbor)
//   out[n, j, :] = w[spatial_neighbors[n, j], :]
//
// Roofline: ~6.4 GFLOP total vs ~290 MB of traffic (256 MB output) =>
// bandwidth-bound on the output store. Compute is free, so we keep full fp32
// precision using V_WMMA_F32_16X16X4_F32 (exact vs the fp32 reference).
// ---------------------------------------------------------------------------

typedef __attribute__((ext_vector_type(2))) float v2f;
typedef __attribute__((ext_vector_type(8))) float v8f;

#define NBN 16384
#define KNBR 16
#define DM 256
#define LDS_STRIDE 260   // 256 + 4 pad: row m -> bank (4m + k) & 63, conflict-free

// One wave computes a 16x16 tile of  OUT = A(16x256) @ W^T(256x16) + bias.
// A lives in LDS (stride LDS_STRIDE); W rows (= output columns) read from
// global, served out of L2 (weights are 256 KB, reused by all 1024 blocks).
//
// V_WMMA_F32_16X16X4_F32 VGPR layouts (ISA 7.12.2):
//   A 16x4 : lane&15 = M ; lane>=16 -> K+=2 ; v[0]=K, v[1]=K+1
//   B 4x16 : lane&15 = N ; lane>=16 -> K+=2 ; v[0]=K, v[1]=K+1
//   D 16x16: VGPR r, lanes 0-15 -> M=r ; lanes 16-31 -> M=r+8 ; N=lane&15
__device__ __forceinline__ v8f tile_gemm(const float* __restrict__ lds_a,
                                         const float* __restrict__ W,
                                         const float* __restrict__ bias,
                                         int nbase, int lane) {
  const int mn    = lane & 15;            // M for A loads, N for B loads
  const int khalf = (lane >> 4) << 1;     // 0 for lanes 0-15, 2 for 16-31
  const int nrow  = nbase + mn;           // W row == output column
  const float* __restrict__ wrow = W + (size_t)nrow * DM;
  const float* __restrict__ arow = lds_a + mn * LDS_STRIDE;

  v8f acc = {};
  #pragma unroll 8
  for (int k0 = 0; k0 < DM; k0 += 4) {
    v2f a, b;
    a.x = arow[k0 + khalf];
    a.y = arow[k0 + khalf + 1];
    const float2 wv = *(const float2*)(wrow + k0 + khalf);
    b.x = wv.x;
    b.y = wv.y;
    // 8 args: (neg_a, A, neg_b, B, c_mod, C, reuse_a, reuse_b)
    acc = __builtin_amdgcn_wmma_f32_16x16x4_f32(
        false, a, false, b, (short)0, acc, false, false);
  }
  const float bv = bias[nrow];
  #pragma unroll
  for (int i = 0; i < 8; ++i) acc[i] += bv;
  return acc;
}

__device__ __forceinline__ void store_tile_lds(float* __restrict__ lds_o,
                                               v8f c, int nbase, int lane) {
  const int n     = nbase + (lane & 15);
  const int mbase = (lane >> 4) << 3;     // 0 or 8
  #pragma unroll
  for (int r = 0; r < 8; ++r)
    lds_o[(mbase + r) * LDS_STRIDE + n] = c[r];
}

// Grid: NBN/16 blocks x 512 threads (16 wave32). Block handles 16 node rows;
// wave `wid` owns output columns [wid*16, wid*16+16). Chained GEMMs ping-pong
// the 16x256 activation between two LDS buffers.
__global__ __launch_bounds__(512)
void proj_chain_kernel(const float* __restrict__ query,
                       const float* __restrict__ Wq, const float* __restrict__ bq,
                       const float* __restrict__ Wv, const float* __restrict__ bv,
                       const float* __restrict__ Wo, const float* __restrict__ bo,
                       float* __restrict__ w_out) {
  __shared__ float ldsA[16 * LDS_STRIDE];
  __shared__ float ldsB[16 * LDS_STRIDE];

  const int tid   = threadIdx.x;
  const int lane  = tid & 31;
  const int wid   = tid >> 5;             // 0..15
  const int nbase = wid << 4;
  const size_t row0 = (size_t)blockIdx.x << 4;

  // Cooperative load: 16x256 fp32 query tile, float4 per access.
  for (int i = tid; i < 16 * (DM / 4); i += 512) {
    const int rr = i >> 6;
    const int cc = (i & 63) << 2;
    const float4 qv = *(const float4*)(query + (row0 + rr) * DM + cc);
    float* dst = &ldsA[rr * LDS_STRIDE + cc];
    dst[0] = qv.x; dst[1] = qv.y; dst[2] = qv.z; dst[3] = qv.w;
  }
  __syncthreads();

  v8f c = tile_gemm(ldsA, Wq, bq, nbase, lane);        // q = query@Wq^T + bq
  store_tile_lds(ldsB, c, nbase, lane);
  __syncthreads();

  c = tile_gemm(ldsB, Wv, bv, nbase, lane);            // v = q@Wv^T + bv
  store_tile_lds(ldsA, c, nbase, lane);
  __syncthreads();

  c = tile_gemm(ldsA, Wo, bo, nbase, lane);            // w = v@Wo^T + bo
  {
    const int n     = nbase + (lane & 15);
    const int mbase = (lane >> 4) << 3;
    #pragma unroll
    for (int r = 0; r < 8; ++r)
      w_out[(row0 + mbase + r) * DM + n] = c[r];
  }
}

// out[row, :] = w[idx[row], :]  for row in [0, NBN*KNBR); one float4 per thread.
__global__ __launch_bounds__(256)
void gather_kernel(const float* __restrict__ w,
                   const int* __restrict__ idx,
                   float* __restrict__ out) {
  const size_t t  = (size_t)blockIdx.x * 256 + threadIdx.x;
  const size_t row = t >> 6;              // 64 float4 per 256-float row
  const int    c4  = (int)(t & 63) << 2;
  const int    src = idx[row];
  const float4 v = *(const float4*)(w + (size_t)src * DM + c4);
  *(float4*)(out + row * DM + c4) = v;
}

extern "C" void kernel_launch(void* const* d_in, const int* in_sizes, int n_in,
                              void* d_out, int out_size, void* d_ws, size_t ws_size,
                              hipStream_t stream) {
  const float* query = (const float*)d_in[0];
  const int*   nbr   = (const int*)d_in[1];
  // d_in[2] = mask : numerically dead (softmax over singleton axis == 1)
  const float* Wq = (const float*)d_in[3];
  const float* bq = (const float*)d_in[4];
  // d_in[5], d_in[6] = Wk, bk : numerically dead (only feed the dead scores)
  const float* Wv = (const float*)d_in[7];
  const float* bv = (const float*)d_in[8];
  const float* Wo = (const float*)d_in[9];
  const float* bo = (const float*)d_in[10];

  float* out = (float*)d_out;             // [NBN, KNBR, DM]
  float* w   = (float*)d_ws;              // [NBN, DM] fp32 = 16 MB scratch

  proj_chain_kernel<<<NBN / 16, 512, 0, stream>>>(query, Wq, bq, Wv, bv, Wo, bo, w);

  const int gather_blocks = (NBN * KNBR * (DM / 4)) / 256;  // 65536
  gather_kernel<<<gather_blocks, 256, 0, stream>>>(w, nbr, out);
}